// GRUWeatherForecaster_2044404433467
// MI455X (gfx1250) — compile-verified
//
#include <hip/hip_runtime.h>

typedef float v2f __attribute__((ext_vector_type(2)));
typedef float v8f __attribute__((ext_vector_type(8)));

#define LOG2E 1.4426950408889634f

// lane swap across the two 16-lane halves (SWAPX16): xor_mask=0x10, and_mask=0x1f
__device__ __forceinline__ float swz16(float v) {
  return __int_as_float(__builtin_amdgcn_ds_swizzle(__float_as_int(v), 0x401f));
}
__device__ __forceinline__ float fast_sigmoid(float x) {
  float e = __builtin_amdgcn_exp2f(-LOG2E * x);
  return __builtin_amdgcn_rcpf(1.0f + e);
}
__device__ __forceinline__ float fast_tanh(float x) {
  float e = __builtin_amdgcn_exp2f(-2.0f * LOG2E * x);
  return fmaf(2.0f, __builtin_amdgcn_rcpf(1.0f + e), -1.0f);
}

// One wave processes 16 sequences. State S[j,b] (hidden row j, batch col b) lives
// in WMMA C/D layout: VGPR v of lane l holds S[v + 8*(l>>4), l&15].
__global__ __launch_bounds__(64) void gru_fused_kernel(
    const float* __restrict__ x, const float* __restrict__ W_ih,
    const float* __restrict__ W_hh, const float* __restrict__ b_ih,
    const float* __restrict__ b_hh, const float* __restrict__ W_fc,
    const float* __restrict__ b_fc, float* __restrict__ out, int T) {
  const int lane = threadIdx.x & 31;
  const int wid  = (blockIdx.x * blockDim.x + threadIdx.x) >> 5;
  const int b0   = wid * 16;          // first sequence of this wave's tile
  const int col  = lane & 15;         // batch column (also A-layout row M)
  const int hi   = lane >> 4;         // which 16-lane half
  const int ks   = hi * 2;            // A-layout K select within a 4-chunk
  const bool lower = (hi == 0);

  // ---- Preload W_hh gate blocks in 16x4 f32 A-layout: per gate, 4 K-chunks ----
  v2f wAr[4], wAz[4], wAn[4];
#pragma unroll
  for (int c = 0; c < 4; ++c) {
    const int kb = 4 * c + ks;
    v2f a;
    a.x = W_hh[(0 + col) * 16 + kb];  a.y = W_hh[(0 + col) * 16 + kb + 1];  wAr[c] = a;
    a.x = W_hh[(16 + col) * 16 + kb]; a.y = W_hh[(16 + col) * 16 + kb + 1]; wAz[c] = a;
    a.x = W_hh[(32 + col) * 16 + kb]; a.y = W_hh[(32 + col) * 16 + kb + 1]; wAn[c] = a;
  }

  // ---- Per-VGPR-row constants in C/D layout (row g = v + 8*hi) ----
  float wihr[8], wihz[8], wihn[8], br[8], bz[8], bihn[8], bhn[8], wfc[8];
#pragma unroll
  for (int v = 0; v < 8; ++v) {
    const int g = v + 8 * hi;
    wihr[v] = W_ih[g]; wihz[v] = W_ih[16 + g]; wihn[v] = W_ih[32 + g];
    br[v]   = b_ih[g] + b_hh[g];                 // r-gate combined bias
    bz[v]   = b_ih[16 + g] + b_hh[16 + g];       // z-gate combined bias
    bihn[v] = b_ih[32 + g];                      // n-gate input-side bias
    bhn[v]  = b_hh[32 + g];                      // n-gate hidden-side bias (inside r*)
    wfc[v]  = W_fc[g];
  }

  v8f s = {};  // h0 = 0

  const float4* __restrict__ xrow =
      reinterpret_cast<const float4*>(x + (size_t)(b0 + col) * (size_t)T);

  for (int t4 = 0; t4 < (T >> 2); ++t4) {
    const float4 xq = xrow[t4];
    float xs[4] = {xq.x, xq.y, xq.z, xq.w};
#pragma unroll
    for (int u = 0; u < 4; ++u) {
      const float xv = xs[u];

      // Build B operand chunks (4x16 each) from state via half-wave swaps.
      const float w0 = swz16(s[0]), w1 = swz16(s[1]), w2 = swz16(s[2]), w3 = swz16(s[3]);
      const float w4 = swz16(s[4]), w5 = swz16(s[5]), w6 = swz16(s[6]), w7 = swz16(s[7]);
      v2f B0, B1, B2, B3;
      B0.x = lower ? s[0] : w2;  B0.y = lower ? s[1] : w3;   // K rows 0..3
      B1.x = lower ? s[4] : w6;  B1.y = lower ? s[5] : w7;   // K rows 4..7
      B2.x = lower ? w0 : s[2];  B2.y = lower ? w1 : s[3];   // K rows 8..11
      B3.x = lower ? w4 : s[6];  B3.y = lower ? w5 : s[7];   // K rows 12..15

      // Three 16x16 matvec tiles, K=16 as 4 chained f32 WMMAs, interleaved so
      // accumulator RAW chains are 3 instructions apart.
      v8f aR = {}, aZ = {}, aN = {};
      aR = __builtin_amdgcn_wmma_f32_16x16x4_f32(false, wAr[0], false, B0, (short)0, aR, false, false);
      aZ = __builtin_amdgcn_wmma_f32_16x16x4_f32(false, wAz[0], false, B0, (short)0, aZ, false, false);
      aN = __builtin_amdgcn_wmma_f32_16x16x4_f32(false, wAn[0], false, B0, (short)0, aN, false, false);
      aR = __builtin_amdgcn_wmma_f32_16x16x4_f32(false, wAr[1], false, B1, (short)0, aR, false, false);
      aZ = __builtin_amdgcn_wmma_f32_16x16x4_f32(false, wAz[1], false, B1, (short)0, aZ, false, false);
      aN = __builtin_amdgcn_wmma_f32_16x16x4_f32(false, wAn[1], false, B1, (short)0, aN, false, false);
      aR = __builtin_amdgcn_wmma_f32_16x16x4_f32(false, wAr[2], false, B2, (short)0, aR, false, false);
      aZ = __builtin_amdgcn_wmma_f32_16x16x4_f32(false, wAz[2], false, B2, (short)0, aZ, false, false);
      aN = __builtin_amdgcn_wmma_f32_16x16x4_f32(false, wAn[2], false, B2, (short)0, aN, false, false);
      aR = __builtin_amdgcn_wmma_f32_16x16x4_f32(false, wAr[3], false, B3, (short)0, aR, false, false);
      aZ = __builtin_amdgcn_wmma_f32_16x16x4_f32(false, wAz[3], false, B3, (short)0, aZ, false, false);
      aN = __builtin_amdgcn_wmma_f32_16x16x4_f32(false, wAn[3], false, B3, (short)0, aN, false, false);

      // Gates + state update, all in C/D layout.
#pragma unroll
      for (int v = 0; v < 8; ++v) {
        const float r = fast_sigmoid(aR[v] + fmaf(xv, wihr[v], br[v]));
        const float z = fast_sigmoid(aZ[v] + fmaf(xv, wihz[v], bz[v]));
        const float n = fast_tanh(fmaf(r, aN[v] + bhn[v], fmaf(xv, wihn[v], bihn[v])));
        s[v] = fmaf(z, s[v] - n, n);   // (1-z)*n + z*h
      }
    }
  }

  // ---- out[b] = sum_j h[b,j] * W_fc[0,j] + b_fc ----
  float p = 0.0f;
#pragma unroll
  for (int v = 0; v < 8; ++v) p = fmaf(s[v], wfc[v], p);
  const float q = p + swz16(p);        // combine the two half-wave partials
  if (lane < 16) out[b0 + lane] = q + b_fc[0];
}

extern "C" void kernel_launch(void* const* d_in, const int* in_sizes, int n_in,
                              void* d_out, int out_size, void* d_ws, size_t ws_size,
                              hipStream_t stream) {
  const float* x    = (const float*)d_in[0];
  const float* W_ih = (const float*)d_in[1];
  const float* W_hh = (const float*)d_in[2];
  const float* b_ih = (const float*)d_in[3];
  const float* b_hh = (const float*)d_in[4];
  const float* W_fc = (const float*)d_in[5];
  const float* b_fc = (const float*)d_in[6];
  float* out = (float*)d_out;

  const int B = out_size;              // 16384
  const int T = in_sizes[0] / B;       // 2048  (I = 1)
  const int waves = B / 16;            // 1024 independent waves
  const int block = 64;                // 2 waves/block -> 512 blocks, max WGP spread
  const int grid  = waves / (block / 32);

  hipLaunchKernelGGL(gru_fused_kernel, dim3(grid), dim3(block), 0, stream,
                     x, W_ih, W_hh, b_ih, b_hh, W_fc, b_fc, out, T);
}